// MultiHeadAttention_34703335751902
// MI455X (gfx1250) — compile-verified
//
#include <hip/hip_runtime.h>
#include <hip/hip_bf16.h>
#include <math.h>

typedef __attribute__((ext_vector_type(2))) float v2f;
typedef __attribute__((ext_vector_type(4))) float v4f;
typedef __attribute__((ext_vector_type(8))) float v8f;

#define B_  4
#define H_  16
#define T_  1024
#define D_  64
#define NB  33            // 2*MAX_REL+1 relative buckets
#define SCALE_ 0.125f     // 1/sqrt(64)
#define WAVES 8
#define RPW   16          // rows (t) per wave
#define RPWG  (WAVES * RPW)

__global__ __launch_bounds__(256, 1)
void relattn_fused_kernel(const float* __restrict__ q,
                          const float* __restrict__ k,
                          const float* __restrict__ v,
                          const float* __restrict__ emb_k,
                          const float* __restrict__ emb_v,
                          float* __restrict__ out,
                          float* __restrict__ attn)
{
    __shared__ float ek_lds[NB][D_];               // 8.4 KB
    __shared__ float ev_lds[NB][D_];               // 8.4 KB
    __shared__ float p_lds[WAVES][RPW][NB + 1];    // rel-k bias per row   17 KB
    __shared__ float w_lds[WAVES][RPW][NB + 1];    // bucket sums of attn  17 KB
    __shared__ float at_lds[WAVES][2][16][17];     // ping-pong attn tile  17 KB

    const int tid  = threadIdx.x;
    const int wave = tid >> 5;
    const int lane = tid & 31;
    const int lm   = lane & 15;   // t_local in the transposed tile
    const int hi   = lane >> 4;   // half-wave select

    const int bh   = blockIdx.y;
    const int row0 = blockIdx.x * RPWG + wave * RPW;  // first t row of wave

    const float* qb = q + (size_t)bh * T_ * D_;
    const float* kb = k + (size_t)bh * T_ * D_;
    const float* vb = v + (size_t)bh * T_ * D_;
    float* outb  = out  + (size_t)bh * T_ * D_;
    float* attnb = attn + (size_t)bh * T_ * T_;

    // ---- stage embedding tables into LDS, zero bucket accumulators ----
    for (int i = tid; i < NB * D_; i += 256) {
        ek_lds[i / D_][i % D_] = emb_k[i];
        ev_lds[i / D_][i % D_] = emb_v[i];
    }
    for (int i = lane; i < RPW * (NB + 1); i += 32)
        w_lds[wave][i / (NB + 1)][i % (NB + 1)] = 0.0f;
    __syncthreads();

    // ---- p[t][j] = Q[t,:] . emb_k[j,:]  (content-position bias) ----
    for (int idx = lane; idx < RPW * NB; idx += 32) {
        const int r = idx / NB, j = idx % NB;
        const float* qr = qb + (size_t)(row0 + r) * D_;
        float acc = 0.f;
        #pragma unroll 8
        for (int d = 0; d < D_; ++d) acc += qr[d] * ek_lds[j][d];
        p_lds[wave][r][j] = acc;
    }
    __syncthreads();

    // per-lane clipped-edge biases (cover all tiles with |s0-row0| >= 32)
    const float pE0  = p_lds[wave][lm][0];
    const float pE32 = p_lds[wave][lm][NB - 1];

    // ---- Q strip as WMMA *B* operand: b.x=Q[t0+n, 4c+2hi], b.y=.x+1 ----
    v2f qv[16];
    {
        const float* qr = qb + (size_t)(row0 + lm) * D_ + 2 * hi;
        #pragma unroll
        for (int c = 0; c < 16; ++c) qv[c] = *(const v2f*)(qr + 4 * c);
    }

    // Transposed logits tile S^T[s,t] = K . Q^T for s in [s0,s0+16).
    // D-tile element (vgpr r, lane): s = s0 + r + 8*hi, t = row0 + lm.
    auto qk_tile = [&](int s0) -> v8f {
        v2f kv[16];
        const float* kr = kb + (size_t)(s0 + lm) * D_ + 2 * hi;  // A layout
        #pragma unroll
        for (int c = 0; c < 16; ++c) kv[c] = *(const v2f*)(kr + 4 * c);
        v8f acc = {0.f, 0.f, 0.f, 0.f, 0.f, 0.f, 0.f, 0.f};
        #pragma unroll
        for (int c = 0; c < 16; ++c)
            acc = __builtin_amdgcn_wmma_f32_16x16x4_f32(
                false, kv[c], false, qv[c], (short)0, acc, false, false);
        return acc;
    };

    // bias + scale; s0-row0 is wave-uniform -> force a *scalar* branch
    auto apply_bias = [&](int s0, const v8f& acc, float tl[8]) {
        const int dd = __builtin_amdgcn_readfirstlane(s0 - row0);
        if (dd <= -32) {                       // all clip to bucket 0
            #pragma unroll
            for (int r = 0; r < 8; ++r) tl[r] = (acc[r] + pE0) * SCALE_;
        } else if (dd >= 32) {                 // all clip to bucket 32
            #pragma unroll
            for (int r = 0; r < 8; ++r) tl[r] = (acc[r] + pE32) * SCALE_;
        } else {                               // diagonal band: 3 tiles/wave
            #pragma unroll
            for (int r = 0; r < 8; ++r) {
                const int delta = dd + r + 8 * hi - lm;
                const int j     = min(max(delta, -16), 16) + 16;
                tl[r] = (acc[r] + p_lds[wave][lm][j]) * SCALE_;
            }
        }
    };

    // ---- pass 1: per-lane online row max / sum-exp (no shuffles in loop) ----
    float mrun = -INFINITY, srun = 0.f;
    for (int s0 = 0; s0 < T_; s0 += 16) {
        const v8f acc = qk_tile(s0);
        float tl[8];
        apply_bias(s0, acc, tl);
        float tmax = tl[0];
        #pragma unroll
        for (int r = 1; r < 8; ++r) tmax = fmaxf(tmax, tl[r]);
        const float mnew = fmaxf(mrun, tmax);
        float es = 0.f;
        #pragma unroll
        for (int r = 0; r < 8; ++r) es += __expf(tl[r] - mnew);
        srun = srun * __expf(mrun - mnew) + es;
        mrun = mnew;
    }
    // combine the two half-wave partials for row t (lanes n and n+16)
    {
        const float mo   = __shfl_xor(mrun, 16);
        const float mall = fmaxf(mrun, mo);
        const float sadj = srun * __expf(mrun - mall);
        srun = sadj + __shfl_xor(sadj, 16);
        mrun = mall;
    }
    const float rinv = 1.0f / srun;

    // ---- pass 2: attn out + attn@V + bucket sums ----
    v8f oacc[4];
    #pragma unroll
    for (int dt = 0; dt < 4; ++dt)
        oacc[dt] = (v8f){0.f, 0.f, 0.f, 0.f, 0.f, 0.f, 0.f, 0.f};

    float wlo = 0.f, whi = 0.f;   // register-accumulated edge buckets

    for (int s0 = 0; s0 < T_; s0 += 16) {
        const v8f acc = qk_tile(s0);
        float tl[8];
        apply_bias(s0, acc, tl);
        float at[8];
        #pragma unroll
        for (int r = 0; r < 8; ++r)
            at[r] = __expf(tl[r] - mrun) * rinv;

        float (*atb)[17] = at_lds[wave][(s0 >> 4) & 1];   // ping-pong buffer

        // stash transposed tile in LDS; bucket accumulation
        #pragma unroll
        for (int r = 0; r < 8; ++r)
            atb[r + 8 * hi][lm] = at[r];

        const int dd = __builtin_amdgcn_readfirstlane(s0 - row0);
        if (dd <= -32) {
            #pragma unroll
            for (int r = 0; r < 8; ++r) wlo += at[r];
        } else if (dd >= 32) {
            #pragma unroll
            for (int r = 0; r < 8; ++r) whi += at[r];
        } else {
            #pragma unroll
            for (int r = 0; r < 8; ++r) {
                const int delta = dd + r + 8 * hi - lm;
                const int j     = min(max(delta, -16), 16) + 16;
                atomicAdd(&w_lds[wave][lm][j], at[r]);   // ds_add_f32
            }
        }
        __builtin_amdgcn_wave_barrier();
        __asm__ volatile("s_wait_dscnt 0" ::: "memory");  // store->read visibility

        // coalesced attn write: lane -> 32B contiguous of one row
        {
            const int tw = lane >> 1;
            const int sw = (lane & 1) * 8;
            v4f x0, x1;
            #pragma unroll
            for (int i = 0; i < 4; ++i) {
                x0[i] = atb[sw + i][tw];
                x1[i] = atb[sw + 4 + i][tw];
            }
            float* dst = attnb + (size_t)(row0 + tw) * T_ + s0 + sw;
            *(v4f*)dst       = x0;
            *((v4f*)dst + 1) = x1;
        }

        // reload attn tile in A-layout [t][s], multiply V tile (B-layout)
        v2f aa[4];
        #pragma unroll
        for (int sk = 0; sk < 4; ++sk) {
            aa[sk].x = atb[4 * sk + 2 * hi][lm];
            aa[sk].y = atb[4 * sk + 2 * hi + 1][lm];
        }
        #pragma unroll
        for (int dt = 0; dt < 4; ++dt) {
            #pragma unroll
            for (int sk = 0; sk < 4; ++sk) {
                const float* vr = vb + (size_t)(s0 + 4 * sk + 2 * hi) * D_ + dt * 16 + lm;
                v2f bv;
                bv.x = vr[0];
                bv.y = vr[D_];
                oacc[dt] = __builtin_amdgcn_wmma_f32_16x16x4_f32(
                    false, aa[sk], false, bv, (short)0, oacc[dt], false, false);
            }
        }
        __builtin_amdgcn_wave_barrier();   // keep next-iter stores behind reads
    }

    // flush register-accumulated edge buckets (both half-waves contribute)
    atomicAdd(&w_lds[wave][lm][0],      wlo);
    atomicAdd(&w_lds[wave][lm][NB - 1], whi);
    __builtin_amdgcn_wave_barrier();
    __asm__ volatile("s_wait_dscnt 0" ::: "memory");

    // ---- epilogue: out += w . emb_v, store (j-loop rolled: small icache) ----
    #pragma unroll
    for (int dt = 0; dt < 4; ++dt) {
        float accs[8];
        #pragma unroll
        for (int r = 0; r < 8; ++r) accs[r] = oacc[dt][r];
        const int d = dt * 16 + lm;
        #pragma unroll 1
        for (int j = 0; j < NB; ++j) {
            const float evv = ev_lds[j][d];
            #pragma unroll
            for (int r = 0; r < 8; ++r)
                accs[r] += w_lds[wave][r + 8 * hi][j] * evv;
        }
        #pragma unroll
        for (int r = 0; r < 8; ++r) {
            const int rowl = r + 8 * hi;
            outb[(size_t)(row0 + rowl) * D_ + d] = accs[r];
        }
    }
}

extern "C" void kernel_launch(void* const* d_in, const int* in_sizes, int n_in,
                              void* d_out, int out_size, void* d_ws, size_t ws_size,
                              hipStream_t stream) {
    (void)in_sizes; (void)n_in; (void)d_ws; (void)ws_size; (void)out_size;
    const float* q     = (const float*)d_in[0];
    const float* k     = (const float*)d_in[1];
    const float* v     = (const float*)d_in[2];
    const float* emb_k = (const float*)d_in[3];
    const float* emb_v = (const float*)d_in[4];

    float* out  = (float*)d_out;                        // [B,H,T,D]
    float* attn = out + (size_t)B_ * H_ * T_ * D_;      // [B,H,T,T]

    dim3 grid(T_ / RPWG, B_ * H_);   // 8 row-strips x 64 (b,h)
    relattn_fused_kernel<<<grid, 256, 0, stream>>>(q, k, v, emb_k, emb_v, out, attn);
}